// EdgeConvGNN_21096879358623
// MI455X (gfx1250) — compile-verified
//
#include <hip/hip_runtime.h>

typedef float v2f __attribute__((ext_vector_type(2)));
typedef float v8f __attribute__((ext_vector_type(8)));

#define DH 128

#define WMMA_F32(a, b, c) \
    __builtin_amdgcn_wmma_f32_16x16x4_f32(false, (a), false, (b), (short)0, (c), false, false)

// async global->LDS copy of 16 bytes per lane (512B per wave), ASYNCcnt-tracked
__device__ __forceinline__ void async_ld_b128(const void* gptr, void* lptr) {
    unsigned laddr = (unsigned)(uintptr_t)lptr;                 // addr[31:0] == LDS offset
    unsigned long long gaddr = (unsigned long long)(uintptr_t)gptr;
    asm volatile("global_load_async_to_lds_b128 %0, %1, off"
                 :: "v"(laddr), "v"(gaddr) : "memory");
}
__device__ __forceinline__ void async_wait0() {
    asm volatile("s_wait_asynccnt 0" ::: "memory");
}

// ---------------- degree / rsqrt ----------------
__global__ __launch_bounds__(256) void k_deg(const int* __restrict__ ldst,
                                             float* __restrict__ deg, long long E) {
    long long i = blockIdx.x * 256LL + threadIdx.x;
    if (i < E) atomicAdd(&deg[ldst[i]], 1.0f);
}

__global__ __launch_bounds__(256) void k_rsqrt(float* __restrict__ deg, int M) {
    int i = blockIdx.x * 256 + threadIdx.x;
    if (i < M) deg[i] = rsqrtf(deg[i] + 1.0f);   // +1 self loop
}

// ---------------- GEMM1: relu(concat(x[src],x[dst]) @ W1 + b1) ----------------
// block = 32 edges x 128 outputs; 8 waves; wave w owns N-tile [16w,16w+16),
// 2 M-subtiles per wave (B fragment reused across both WMMAs).
__global__ __launch_bounds__(256) void k_gemm_edge(
    const float* __restrict__ x, const int* __restrict__ gsrc, const int* __restrict__ gdst,
    const float* __restrict__ W1, const float* __restrict__ b1,
    float* __restrict__ h, int M)
{
    __shared__ float As[32 * 260];               // 32 x 256, pitch 260 floats (bank-safe)
    const int m0 = blockIdx.x * 32;

    // async-stage gathered tile: 2048 x 16B chunks, 8 per thread
    #pragma unroll
    for (int i = 0; i < 8; ++i) {
        int c   = threadIdx.x + i * 256;
        int row = c >> 6;                        // 64 float4-chunks per 256-float row
        int c4  = c & 63;
        int e = m0 + row; if (e >= M) e = M - 1;
        int node = (c4 < 32) ? gsrc[e] : gdst[e];
        const float* gp = x + (size_t)node * DH + (size_t)(c4 & 31) * 4;
        async_ld_b128(gp, &As[row * 260 + c4 * 4]);
    }
    async_wait0();
    __syncthreads();

    const int wv   = threadIdx.x >> 5;
    const int lane = threadIdx.x & 31;
    const int half = lane >> 4;
    const int l    = lane & 15;
    const int n0   = wv * 16;
    const float* wcol = W1 + n0 + l;             // W1[k][n] row-major -> column walk

    v8f acc0 = {}, acc1 = {};
    #pragma unroll 4
    for (int k0 = 0; k0 < 2 * DH; k0 += 4) {
        const int ka = k0 + 2 * half;            // A/B VGPR0 = K{0,2}, VGPR1 = K{1,3}
        v2f bv;
        bv.x = wcol[(size_t)ka * DH];
        bv.y = wcol[(size_t)(ka + 1) * DH];
        v2f a0, a1;
        a0.x = As[l * 260 + ka];        a0.y = As[l * 260 + ka + 1];
        a1.x = As[(l + 16) * 260 + ka]; a1.y = As[(l + 16) * 260 + ka + 1];
        acc0 = WMMA_F32(a0, bv, acc0);
        acc1 = WMMA_F32(a1, bv, acc1);
    }

    const float bias = b1[n0 + l];
    if (m0 + 32 <= M) {                          // fast path: single base + imm offsets
        float* hp = h + (size_t)(m0 + half * 8) * DH + n0 + l;
        #pragma unroll
        for (int r = 0; r < 8; ++r) {            // C/D: VGPR r = rows r / r+8
            hp[r * DH]             = fmaxf(acc0[r] + bias, 0.0f);
            hp[(r + 16) * DH]      = fmaxf(acc1[r] + bias, 0.0f);
        }
    } else {
        #pragma unroll
        for (int r = 0; r < 8; ++r) {
            int row0 = m0 + r + half * 8;
            if (row0 < M)      h[(size_t)row0 * DH + n0 + l]        = fmaxf(acc0[r] + bias, 0.0f);
            if (row0 + 16 < M) h[(size_t)(row0 + 16) * DH + n0 + l] = fmaxf(acc1[r] + bias, 0.0f);
        }
    }
}

// ---------------- GEMM2: u = (h @ Wc) * dinv[row] ----------------
// block = 64 rows x 128 outputs; 8 waves; 4 M-subtiles per wave.
__global__ __launch_bounds__(256) void k_gemm_h(
    const float* __restrict__ hin, const float* __restrict__ Wc,
    const float* __restrict__ dinv, float* __restrict__ u, int M)
{
    __shared__ float As[64 * 132];               // 64 x 128, pitch 132 floats (bank-safe)
    const int m0 = blockIdx.x * 64;

    // async-stage contiguous tile: 2048 x 16B chunks, 8 per thread
    #pragma unroll
    for (int i = 0; i < 8; ++i) {
        int c   = threadIdx.x + i * 256;
        int row = c >> 5;                        // 32 float4-chunks per 128-float row
        int c4  = c & 31;
        int gr = m0 + row; if (gr >= M) gr = M - 1;
        const float* gp = hin + (size_t)gr * DH + (size_t)c4 * 4;
        async_ld_b128(gp, &As[row * 132 + c4 * 4]);
    }
    async_wait0();
    __syncthreads();

    const int wv   = threadIdx.x >> 5;
    const int lane = threadIdx.x & 31;
    const int half = lane >> 4;
    const int l    = lane & 15;
    const int n0   = wv * 16;
    const float* wcol = Wc + n0 + l;

    v8f acc0 = {}, acc1 = {}, acc2 = {}, acc3 = {};
    #pragma unroll 4
    for (int k0 = 0; k0 < DH; k0 += 4) {
        const int ka = k0 + 2 * half;
        v2f bv;
        bv.x = wcol[(size_t)ka * DH];
        bv.y = wcol[(size_t)(ka + 1) * DH];
        v2f a0, a1, a2, a3;
        a0.x = As[l * 132 + ka];        a0.y = As[l * 132 + ka + 1];
        a1.x = As[(l + 16) * 132 + ka]; a1.y = As[(l + 16) * 132 + ka + 1];
        a2.x = As[(l + 32) * 132 + ka]; a2.y = As[(l + 32) * 132 + ka + 1];
        a3.x = As[(l + 48) * 132 + ka]; a3.y = As[(l + 48) * 132 + ka + 1];
        acc0 = WMMA_F32(a0, bv, acc0);
        acc1 = WMMA_F32(a1, bv, acc1);
        acc2 = WMMA_F32(a2, bv, acc2);
        acc3 = WMMA_F32(a3, bv, acc3);
    }

    if (m0 + 64 <= M) {                          // fast path: base + imm offsets
        float* up = u + (size_t)(m0 + half * 8) * DH + n0 + l;
        const float* dp = dinv + m0 + half * 8;
        #pragma unroll
        for (int r = 0; r < 8; ++r) {
            up[r * DH]        = acc0[r] * dp[r];
            up[(r + 16) * DH] = acc1[r] * dp[r + 16];
            up[(r + 32) * DH] = acc2[r] * dp[r + 32];
            up[(r + 48) * DH] = acc3[r] * dp[r + 48];
        }
    } else {
        #pragma unroll
        for (int r = 0; r < 8; ++r) {
            int row0 = m0 + r + half * 8;
            if (row0 < M)      u[(size_t)row0 * DH + n0 + l]        = acc0[r] * dinv[row0];
            if (row0 + 16 < M) u[(size_t)(row0 + 16) * DH + n0 + l] = acc1[r] * dinv[row0 + 16];
            if (row0 + 32 < M) u[(size_t)(row0 + 32) * DH + n0 + l] = acc2[r] * dinv[row0 + 32];
            if (row0 + 48 < M) u[(size_t)(row0 + 48) * DH + n0 + l] = acc3[r] * dinv[row0 + 48];
        }
    }
}

// ---------------- scatter-add: agg[dst] += u[src]  (one wave per edge) --------
__global__ __launch_bounds__(256) void k_scatter(
    const float* __restrict__ u, const int* __restrict__ lsrc,
    const int* __restrict__ ldst, float* __restrict__ agg, long long E)
{
    long long gid = blockIdx.x * 256LL + threadIdx.x;
    long long e = gid >> 5;
    if (e >= E) return;
    int q = (int)(gid & 31);                 // lane -> float4 chunk of the 128-row
    // e is wave-uniform: force the index fetches onto the scalar path
    int eu = __builtin_amdgcn_readfirstlane((int)e);
    int s = lsrc[eu], d = ldst[eu];
    float4 v = ((const float4*)(u + (size_t)s * DH))[q];
    float* ap = agg + (size_t)d * DH + q * 4;
    atomicAdd(ap + 0, v.x);
    atomicAdd(ap + 1, v.y);
    atomicAdd(ap + 2, v.z);
    atomicAdd(ap + 3, v.w);
}

// ---------------- combine: h = relu((agg + u) * dinv + b) ----------------
__global__ __launch_bounds__(256) void k_combine(
    const float* __restrict__ agg, const float* __restrict__ u,
    const float* __restrict__ dinv, const float* __restrict__ bc,
    float* __restrict__ hout, int M)
{
    long long i = blockIdx.x * 256LL + threadIdx.x;         // float4 index
    long long tot = (long long)M * (DH / 4);
    if (i >= tot) return;
    int row = (int)(i >> 5);
    int c4  = (int)(i & 31);
    float di = dinv[row];
    float4 a  = ((const float4*)agg)[i];
    float4 uu = ((const float4*)u)[i];
    float4 bb = ((const float4*)bc)[c4];
    float4 o;
    o.x = fmaxf((a.x + uu.x) * di + bb.x, 0.0f);
    o.y = fmaxf((a.y + uu.y) * di + bb.y, 0.0f);
    o.z = fmaxf((a.z + uu.z) * di + bb.z, 0.0f);
    o.w = fmaxf((a.w + uu.w) * di + bb.w, 0.0f);
    ((float4*)hout)[i] = o;
}

// ---------------- output head: out[e] = h[e,:] . W2 + b2 ----------------
__global__ __launch_bounds__(256) void k_out(
    const float* __restrict__ h, const float* __restrict__ W2,
    const float* __restrict__ b2, float* __restrict__ out, int M)
{
    long long gid = blockIdx.x * 256LL + threadIdx.x;
    long long row = gid >> 5;
    int lane = (int)(gid & 31);
    if (row >= M) return;
    const float* hr = h + row * (long long)DH;
    float s = hr[lane] * W2[lane] + hr[lane + 32] * W2[lane + 32]
            + hr[lane + 64] * W2[lane + 64] + hr[lane + 96] * W2[lane + 96];
    #pragma unroll
    for (int off = 16; off > 0; off >>= 1) s += __shfl_xor(s, off, 32);
    if (lane == 0) out[row] = s + b2[0];
}

// ---------------- host launch ----------------
extern "C" void kernel_launch(void* const* d_in, const int* in_sizes, int n_in,
                              void* d_out, int out_size, void* d_ws, size_t ws_size,
                              hipStream_t stream)
{
    const float* x   = (const float*)d_in[0];
    const int* g_ei  = (const int*)d_in[1];
    const int* lg_ei = (const int*)d_in[2];
    const float* W1  = (const float*)d_in[3];
    const float* b1  = (const float*)d_in[4];
    const float* Wc0 = (const float*)d_in[5];
    const float* bc0 = (const float*)d_in[6];
    const float* Wc1 = (const float*)d_in[7];
    const float* bc1 = (const float*)d_in[8];
    const float* W2  = (const float*)d_in[9];
    const float* b2  = (const float*)d_in[10];

    const int M       = in_sizes[1] / 2;       // E_G: number of line-graph nodes
    const long long E = in_sizes[2] / 2;       // E_LG
    const int* gsrc = g_ei;
    const int* gdst = g_ei + M;
    const int* lsrc = lg_ei;
    const int* ldst = lg_ei + E;

    size_t per = ((size_t)M * DH * sizeof(float) + 255) & ~(size_t)255;
    char* ws = (char*)d_ws;
    float* bufH = (float*)(ws);                // current h
    float* bufU = (float*)(ws + per);          // u = (h@Wc)*dinv
    float* bufA = (float*)(ws + 2 * per);      // scatter accumulator
    float* dinv = (float*)(ws + 3 * per);      // deg -> rsqrt in place

    // degrees (shared by both GCN layers)
    hipMemsetAsync(dinv, 0, (size_t)M * sizeof(float), stream);
    k_deg<<<(int)((E + 255) / 256), 256, 0, stream>>>(ldst, dinv, E);
    k_rsqrt<<<(M + 255) / 256, 256, 0, stream>>>(dinv, M);

    k_gemm_edge<<<(M + 31) / 32, 256, 0, stream>>>(x, gsrc, gdst, W1, b1, bufH, M);

    const float* Wcs[2] = {Wc0, Wc1};
    const float* bcs[2] = {bc0, bc1};
    for (int layer = 0; layer < 2; ++layer) {
        k_gemm_h<<<(M + 63) / 64, 256, 0, stream>>>(bufH, Wcs[layer], dinv, bufU, M);
        hipMemsetAsync(bufA, 0, (size_t)M * DH * sizeof(float), stream);
        long long sthreads = E * 32;
        k_scatter<<<(int)((sthreads + 255) / 256), 256, 0, stream>>>(bufU, lsrc, ldst, bufA, E);
        long long c4 = (long long)M * (DH / 4);
        k_combine<<<(int)((c4 + 255) / 256), 256, 0, stream>>>(bufA, bufU, dinv, bcs[layer], bufH, M);
    }

    long long othreads = (long long)M * 32;
    k_out<<<(int)((othreads + 255) / 256), 256, 0, stream>>>(bufH, W2, b2, (float*)d_out, M);
}